// GCN_13331578486814
// MI455X (gfx1250) — compile-verified
//
#include <hip/hip_runtime.h>
#include <hip/hip_bf16.h>

typedef __attribute__((ext_vector_type(2))) float v2f;
typedef __attribute__((ext_vector_type(8))) float v8f;

#define D_FEAT 128
#define TILE_M 64
#define LDS_STRIDE 132   // multiple of 4 (16B-aligned float4 rows); mod-64-banks = 4

// ---------------------------------------------------------------------------
// Degree with self-loop: deg = 1 + indegree(dst)  (exact, integer atomics)
// ---------------------------------------------------------------------------
__global__ void k_deg_init(int* deg, int n) {
    int i = blockIdx.x * blockDim.x + threadIdx.x;
    if (i < n) deg[i] = 1;
}

__global__ void k_deg_count(const int* __restrict__ dst, int* deg, int e) {
    int i = blockIdx.x * blockDim.x + threadIdx.x;
    if (i < e) atomicAdd(&deg[dst[i]], 1);
}

__global__ void k_dinv(const int* __restrict__ deg, float* __restrict__ dinv, int n) {
    int i = blockIdx.x * blockDim.x + threadIdx.x;
    if (i < n) dinv[i] = rsqrtf((float)deg[i]);
}

// Per-edge normalization coefficient: norm[e] = dinv[src]*dinv[dst]
__global__ void k_edge_norm(const int* __restrict__ src, const int* __restrict__ dst,
                            const float* __restrict__ dinv, float* __restrict__ norm,
                            int e) {
    int i = blockIdx.x * blockDim.x + threadIdx.x;
    if (i < e) norm[i] = dinv[src[i]] * dinv[dst[i]];
}

// ---------------------------------------------------------------------------
// GEMM1 + fused self-loop scaling:
//   H[N,128]    = X @ W1                          (V_WMMA_F32_16X16X4_F32)
//   AGG[N,128]  = H * dinv^2                      (self-loop init of scatter dst)
// Block = 256 threads = 8 waves; block covers TILE_M=64 rows (4 row-tiles),
// wave w owns column tile w. W1 staged transposed in LDS so both A and B
// fragments are contiguous ds_load_b64s.
// ---------------------------------------------------------------------------
__global__ __launch_bounds__(256) void k_gemm1_wmma(
        const float* __restrict__ X, const float* __restrict__ W,
        const float* __restrict__ dinv,
        float* __restrict__ H, float* __restrict__ AGG, int n) {
    __shared__ float sWt[D_FEAT * LDS_STRIDE];   // sWt[c*stride + k] = W[k][c]
    __shared__ float sX [TILE_M * LDS_STRIDE];   // sX[r*stride + c]

    const int tid  = threadIdx.x;
    const int row0 = blockIdx.x * TILE_M;
    const bool fullTile = (row0 + TILE_M) <= n;

    // Stage W transposed: float4 global reads, scalar LDS scatter (once/block).
    for (int i = tid; i < D_FEAT * (D_FEAT / 4); i += 256) {
        int k  = i >> 5;         // 0..127
        int c4 = (i & 31) * 4;   // 0,4,..,124
        float4 v = ((const float4*)W)[i];
        sWt[(c4 + 0) * LDS_STRIDE + k] = v.x;
        sWt[(c4 + 1) * LDS_STRIDE + k] = v.y;
        sWt[(c4 + 2) * LDS_STRIDE + k] = v.z;
        sWt[(c4 + 3) * LDS_STRIDE + k] = v.w;
    }
    // Stage 64-row x tile with float4 (ds_store_b128), zero-padded past N.
    if (fullTile) {
        for (int i = tid; i < TILE_M * (D_FEAT / 4); i += 256) {
            int r  = i >> 5;
            int c4 = (i & 31) * 4;
            float4 v = *(const float4*)(X + (size_t)(row0 + r) * D_FEAT + c4);
            *(float4*)(&sX[r * LDS_STRIDE + c4]) = v;
        }
    } else {
        for (int i = tid; i < TILE_M * (D_FEAT / 4); i += 256) {
            int r  = i >> 5;
            int c4 = (i & 31) * 4;
            int g  = row0 + r;
            float4 v = make_float4(0.f, 0.f, 0.f, 0.f);
            if (g < n) v = *(const float4*)(X + (size_t)g * D_FEAT + c4);
            *(float4*)(&sX[r * LDS_STRIDE + c4]) = v;
        }
    }
    __syncthreads();

    const int wave = tid >> 5;          // 0..7 -> column tile
    const int lane = tid & 31;
    const int half = lane >> 4;         // ISA A/B frag: lanes 16-31 hold K+2,K+3
    const int l16  = lane & 15;
    const int colb = wave * 16;

    const float* bBase = &sWt[(colb + l16) * LDS_STRIDE + 2 * half];

    for (int rt = 0; rt < TILE_M / 16; ++rt) {
        v8f acc = {};
        const float* aBase = &sX[(rt * 16 + l16) * LDS_STRIDE + 2 * half];
        #pragma unroll 8
        for (int k = 0; k < D_FEAT; k += 4) {
            v2f a = *(const v2f*)(aBase + k);   // A[m][k+2h], A[m][k+2h+1]
            v2f b = *(const v2f*)(bBase + k);   // B[k+2h][nc], B[k+2h+1][nc]
            acc = __builtin_amdgcn_wmma_f32_16x16x4_f32(
                false, a, false, b, (short)0, acc, false, false);
        }
        // D layout: VGPR r, lanes 0-15: M=r, lanes 16-31: M=r+8; N = lane&15
        const int mrow0 = row0 + rt * 16 + half * 8;
        if (fullTile) {
            #pragma unroll
            for (int r = 0; r < 8; ++r) {
                int gr = mrow0 + r;
                float d = dinv[gr];
                size_t off = (size_t)gr * D_FEAT + colb + l16;
                H[off]   = acc[r];
                AGG[off] = acc[r] * d * d;
            }
        } else {
            #pragma unroll
            for (int r = 0; r < 8; ++r) {
                int gr = mrow0 + r;
                if (gr < n) {
                    float d = dinv[gr];
                    size_t off = (size_t)gr * D_FEAT + colb + l16;
                    H[off]   = acc[r];
                    AGG[off] = acc[r] * d * d;
                }
            }
        }
    }
}

// Edge scatter, 128 features: 1 feature per thread, 2 edges per 256-block.
__global__ __launch_bounds__(256) void k_edge_agg128(
        const int* __restrict__ src, const int* __restrict__ dst,
        const float* __restrict__ h, const float* __restrict__ norm,
        float* __restrict__ out, int e) {
    size_t idx = (size_t)blockIdx.x * blockDim.x + threadIdx.x;
    int eid = (int)(idx >> 7);
    int f   = (int)(idx & 127);
    if (eid < e) {
        int s = src[eid], d = dst[eid];
        unsafeAtomicAdd(&out[(size_t)d * D_FEAT + f],
                        h[(size_t)s * D_FEAT + f] * norm[eid]);
    }
}

// h2 = relu(agg + b1), in place, float4-streamed
__global__ __launch_bounds__(256) void k_bias_relu(
        float* __restrict__ a, const float* __restrict__ b, int n) {
    size_t i = (size_t)blockIdx.x * blockDim.x + threadIdx.x;     // float4 index
    if (i < (size_t)n * (D_FEAT / 4)) {
        float4 v  = ((const float4*)a)[i];
        float4 bb = ((const float4*)b)[i & 31];
        v.x = fmaxf(v.x + bb.x, 0.0f);
        v.y = fmaxf(v.y + bb.y, 0.0f);
        v.z = fmaxf(v.z + bb.z, 0.0f);
        v.w = fmaxf(v.w + bb.w, 0.0f);
        ((float4*)a)[i] = v;
    }
}

// ---------------------------------------------------------------------------
// GEMM2 + fused self-loop + bias:
//   h3[N,2]  = h2 @ W2
//   out[N,2] = h3*dinv^2 + b2     (initializes scatter destination)
// One wave32 per node, shuffle reduction.
// ---------------------------------------------------------------------------
__global__ __launch_bounds__(256) void k_gemm2(
        const float* __restrict__ h2, const float* __restrict__ W2,
        const float* __restrict__ dinv, const float* __restrict__ b2,
        float* __restrict__ h3, float* __restrict__ out, int n) {
    int gwave = (int)(((size_t)blockIdx.x * blockDim.x + threadIdx.x) >> 5);
    int lane = threadIdx.x & 31;
    if (gwave >= n) return;
    const float* row = h2 + (size_t)gwave * D_FEAT;
    float a0 = 0.0f, a1 = 0.0f;
    #pragma unroll
    for (int f = lane; f < D_FEAT; f += 32) {
        float v = row[f];
        a0 += v * W2[2 * f];
        a1 += v * W2[2 * f + 1];
    }
    #pragma unroll
    for (int off = 16; off > 0; off >>= 1) {
        a0 += __shfl_xor(a0, off, 32);
        a1 += __shfl_xor(a1, off, 32);
    }
    if (lane == 0) {
        float dd = dinv[gwave];
        dd = dd * dd;
        h3[2 * gwave]      = a0;
        h3[2 * gwave + 1]  = a1;
        out[2 * gwave]     = a0 * dd + b2[0];
        out[2 * gwave + 1] = a1 * dd + b2[1];
    }
}

__global__ __launch_bounds__(256) void k_edge_agg2(
        const int* __restrict__ src, const int* __restrict__ dst,
        const float* __restrict__ h3, const float* __restrict__ norm,
        float* __restrict__ out, int e) {
    int eid = blockIdx.x * blockDim.x + threadIdx.x;
    if (eid < e) {
        int s = src[eid], d = dst[eid];
        float nm = norm[eid];
        unsafeAtomicAdd(&out[2 * d],     h3[2 * s]     * nm);
        unsafeAtomicAdd(&out[2 * d + 1], h3[2 * s + 1] * nm);
    }
}

// ---------------------------------------------------------------------------
extern "C" void kernel_launch(void* const* d_in, const int* in_sizes, int n_in,
                              void* d_out, int out_size, void* d_ws, size_t ws_size,
                              hipStream_t stream) {
    const float* x  = (const float*)d_in[0];
    const int*   ei = (const int*)  d_in[1];
    const float* W1 = (const float*)d_in[2];
    const float* b1 = (const float*)d_in[3];
    const float* W2 = (const float*)d_in[4];
    const float* b2 = (const float*)d_in[5];
    float* out = (float*)d_out;

    const int n = in_sizes[0] / D_FEAT;   // N nodes
    const int e = in_sizes[1] / 2;        // E edges
    const int* src = ei;
    const int* dst = ei + e;

    // Workspace layout
    char* w = (char*)d_ws;
    size_t o = 0;
    int* deg = (int*)(w + o);          o += (size_t)n * sizeof(int);
    float* dinv = (float*)(w + o);     o += (size_t)n * sizeof(float);
    float* norm = (float*)(w + o);     o += (size_t)e * sizeof(float);
    o = (o + 511) & ~(size_t)511;
    float* h    = (float*)(w + o);     o += (size_t)n * D_FEAT * sizeof(float);
    float* agg1 = (float*)(w + o);     o += (size_t)n * D_FEAT * sizeof(float); // becomes h2
    float* h3   = (float*)(w + o);

    const int B = 256;
    int gN    = (n + B - 1) / B;
    int gE    = (e + B - 1) / B;
    int gNF4  = (int)(((size_t)n * (D_FEAT / 4) + B - 1) / B);
    int gEF   = (int)(((size_t)e * D_FEAT + B - 1) / B);
    int gGemm = (n + TILE_M - 1) / TILE_M;
    int gWav  = (int)(((size_t)n * 32 + B - 1) / B);

    // degree + normalization
    k_deg_init <<<gN, B, 0, stream>>>(deg, n);
    k_deg_count<<<gE, B, 0, stream>>>(dst, deg, e);
    k_dinv     <<<gN, B, 0, stream>>>(deg, dinv, n);
    k_edge_norm<<<gE, B, 0, stream>>>(src, dst, dinv, norm, e);

    // layer 1: GEMM(+self-loop init) -> edge scatter -> bias+relu
    k_gemm1_wmma <<<gGemm, B, 0, stream>>>(x, W1, dinv, h, agg1, n);
    k_edge_agg128<<<gEF, B, 0, stream>>>(src, dst, h, norm, agg1, e);
    k_bias_relu  <<<gNF4, B, 0, stream>>>(agg1, b1, n);     // agg1 -> h2

    // layer 2: GEMM(+self-loop+bias init) -> edge scatter
    k_gemm2    <<<gWav, B, 0, stream>>>(agg1, W2, dinv, b2, h3, out, n);
    k_edge_agg2<<<gE,  B, 0, stream>>>(src, dst, h3, norm, out, e);
}